// Group_for_all_attribute_30193620091439
// MI455X (gfx1250) — compile-verified
//
#include <hip/hip_runtime.h>

// Problem constants (from reference)
#define B_   8
#define N_   16384
#define A_   7
#define NG   1024
#define GS   32
#define TILE 256   // points per KNN tile

typedef __attribute__((ext_vector_type(2))) float v2f;
typedef __attribute__((ext_vector_type(8))) float v8f;
typedef unsigned long long u64;

// ---- lane shuffles via ds_swizzle (immediate pattern, no address VGPR) ----
// group-of-32 pattern: offset[14:10]=xor_mask, [9:5]=or_mask, [4:0]=and_mask
#define XORP(J)      (((J) << 10) | 0x1f)
#define SWZ32(V, P)  __builtin_amdgcn_ds_swizzle((V), (P))
#define SWZ64(V, P)                                                          \
  ((((u64)(unsigned)SWZ32((int)((V) >> 32), (P))) << 32) |                   \
   (unsigned)SWZ32((int)(V), (P)))

// compare-exchange: keep min if !FLIP, keep max if FLIP (u64 keys are unique)
#define CE(VAR, J, FLIP)                                                     \
  { const u64 _ok = SWZ64(VAR, XORP(J));                                     \
    if ((_ok < (VAR)) != (FLIP)) (VAR) = _ok; }

// max-reduce stage across the wave
#define REDMAX(VAR, J)                                                       \
  { const u64 _ok = SWZ64(VAR, XORP(J)); if (_ok > (VAR)) (VAR) = _ok; }

// monotonic float -> u32 (total order, handles negative values)
__device__ __forceinline__ unsigned fmono(float f) {
  unsigned b = __float_as_uint(f);
  return b ^ ((unsigned)((int)b >> 31) | 0x80000000u);
}
// ascending key: smaller distance first, ties -> smaller index (top_k stable)
__device__ __forceinline__ u64 packMinKey(float d, int idx) {
  return ((u64)fmono(d) << 32) | (unsigned)idx;
}
// descending argmax key: larger distance first, ties -> smaller index
__device__ __forceinline__ u64 packMaxKey(float d, int idx) {
  return ((u64)fmono(d) << 32) | (0xffffffffu - (unsigned)idx);
}

// ---------------------------------------------------------------------------
// Kernel 1: farthest point sampling. One block per batch, 1024 threads.
// Coords + running min-distances live in VGPRs (16 pts/thread); per-round
// argmax = packed-u64 xor-swizzle wave reduction + 32-entry LDS round.
// ---------------------------------------------------------------------------
__global__ __launch_bounds__(1024) void fps_kernel(const float* __restrict__ xyz,
                                                   float* __restrict__ out_idx,
                                                   float* __restrict__ out_attr,
                                                   float* __restrict__ out_coor) {
  const int b    = blockIdx.x;
  const int t    = threadIdx.x;
  const int lane = t & 31;
  const int wave = t >> 5;
  const float* base = xyz + (size_t)b * N_ * A_;

  float px[16], py[16], pz[16], dst[16];
#pragma unroll
  for (int j = 0; j < 16; ++j) {
    const int p = t + j * 1024;          // strided -> coalesced initial load
    px[j]  = base[p * A_ + 0];
    py[j]  = base[p * A_ + 1];
    pz[j]  = base[p * A_ + 2];
    dst[j] = 1e10f;
  }

  __shared__ u64   s_key[32];
  __shared__ float s_cent[3];
  __shared__ int   s_far;

  int far = 0;
  for (int i = 0; i < NG; ++i) {
    if (t == 0) {
      out_idx[b * NG + i] = (float)far;
#pragma unroll
      for (int c = 0; c < A_; ++c) {
        const float a = base[(size_t)far * A_ + c];
        out_attr[((size_t)(b * NG + i)) * A_ + c] = a;
        if (c < 3) { out_coor[((size_t)(b * NG + i)) * 3 + c] = a; s_cent[c] = a; }
      }
    }
    __syncthreads();
    const float cx = s_cent[0], cy = s_cent[1], cz = s_cent[2];

    float bv = -1.0f; int bidx = 0;      // ascending p scan + strict '>' keeps
#pragma unroll                           // first (smallest) index on local ties
    for (int j = 0; j < 16; ++j) {
      const float dx = px[j] - cx, dy = py[j] - cy, dz = pz[j] - cz;
      float d = dx * dx + dy * dy + dz * dz;
      d = fminf(d, dst[j]); dst[j] = d;
      if (d > bv) { bv = d; bidx = t + j * 1024; }
    }
    u64 k = packMaxKey(bv, bidx);
    REDMAX(k, 16) REDMAX(k, 8) REDMAX(k, 4) REDMAX(k, 2) REDMAX(k, 1)
    if (lane == 0) s_key[wave] = k;
    __syncthreads();
    if (wave == 0) {
      k = s_key[lane];
      REDMAX(k, 16) REDMAX(k, 8) REDMAX(k, 4) REDMAX(k, 2) REDMAX(k, 1)
      if (lane == 0) s_far = (int)(0xffffffffu - (unsigned)k);
    }
    __syncthreads();
    far = s_far;
  }
}

// ---------------------------------------------------------------------------
// Kernel 2: KNN (7-D) via V_WMMA_F32_16X16X4_F32 + exact wave-bitonic top-32
// on packed u64 keys, then gather + recenter. Block = 16 waves = 16 centers.
//   d2 = |c|^2 + |x|^2 - 2 * (C x X^T), cross term on the matrix pipe.
// A-matrix (16x4 f32) per ISA: lane m = lane&15, K-pair = 2*(lane>>4);
// B-matrix (4x16 f32) mirrored; C: VGPR r -> M = r + 8*(lane>>4), N = lane&15.
// ---------------------------------------------------------------------------
__global__ __launch_bounds__(512) void knn_group_kernel(const float* __restrict__ xyz,
                                                        const float* __restrict__ cattr,
                                                        float* __restrict__ out_nei) {
  const int b     = blockIdx.y;
  const int ctile = blockIdx.x;          // 0..63, 16 centers each
  const int t     = threadIdx.x;
  const int lane  = t & 31;
  const int w     = t >> 5;              // wave 0..15 == local center id

  __shared__ float sC[16][8];            // centers, K padded 7->8
  __shared__ float sC2[16];              // |c|^2
  __shared__ float sP[8][TILE];          // points transposed [K][n]
  __shared__ float sX2[TILE];            // |x|^2
  __shared__ float sD[16][TILE];         // distance tile

  const float* base = xyz + (size_t)b * N_ * A_;

  if (t < 16 * 8) {
    const int m = t >> 3, a = t & 7;
    sC[m][a] = (a < A_) ? cattr[((size_t)(b * NG + ctile * 16 + m)) * A_ + a] : 0.0f;
  }
  __syncthreads();
  if (t < 16) {
    float s = 0.0f;
#pragma unroll
    for (int a = 0; a < A_; ++a) s += sC[t][a] * sC[t][a];
    sC2[t] = s;
  }
  __syncthreads();

  // A operands (held for the whole kernel)
  const int am = lane & 15;
  const int k0 = (lane >> 4) << 1;
  v2f A0, A1;
  A0.x = sC[am][k0];     A0.y = sC[am][k0 + 1];
  A1.x = sC[am][k0 + 4]; A1.y = sC[am][k0 + 5];

  // top-32 state: keys sorted ascending across lanes of the wave
  u64 bk = packMinKey(3e38f, 0);

  for (int tile0 = 0; tile0 < N_; tile0 += TILE) {
    __syncthreads();                      // sP/sD reuse vs previous selection
    for (int e = t; e < 8 * TILE; e += 512) {
      const int a = e >> 8;               // TILE == 256
      const int p = e & (TILE - 1);
      sP[a][p] = (a < A_) ? base[(size_t)(tile0 + p) * A_ + a] : 0.0f;
    }
    __syncthreads();
    if (t < TILE) {
      float s = 0.0f;
#pragma unroll
      for (int a = 0; a < A_; ++a) { const float v = sP[a][t]; s += v * v; }
      sX2[t] = s;
    }
    // prefetch next point tile into cache while we compute/select this one
    if (tile0 + TILE < N_)
      __builtin_prefetch(&base[(size_t)(tile0 + TILE + (t & (TILE - 1))) * A_], 0, 1);
    __syncthreads();

    // ---- phase 1: wave w computes columns [w*16, w*16+16) via WMMA ----
    {
      const int n0 = w * 16;
      const int n  = n0 + (lane & 15);
      v2f B0, B1;
      B0.x = sP[k0][n];     B0.y = sP[k0 + 1][n];
      B1.x = sP[k0 + 4][n]; B1.y = sP[k0 + 5][n];
      v8f acc = {};
      acc = __builtin_amdgcn_wmma_f32_16x16x4_f32(false, A0, false, B0,
                                                  (short)0, acc, false, false);
      acc = __builtin_amdgcn_wmma_f32_16x16x4_f32(false, A1, false, B1,
                                                  (short)0, acc, false, false);
      const int mbase = (lane >> 4) << 3;
      const int ncol  = n0 + (lane & 15);
#pragma unroll
      for (int r = 0; r < 8; ++r) {
        const int m = mbase + r;
        sD[m][ncol] = sC2[m] + sX2[ncol] - 2.0f * acc[r];
      }
    }
    __syncthreads();

    // ---- phase 2: wave w folds this tile's row w into its top-32 ----
    for (int bj = 0; bj < TILE; bj += 32) {
      u64 ck = packMinKey(sD[w][bj + lane], tile0 + bj + lane);
      // prune against current worst (lane 31 of ascending best list)
      const u64 kmax =
          (((u64)(unsigned)__builtin_amdgcn_readlane((int)(bk >> 32), 31)) << 32) |
          (unsigned)__builtin_amdgcn_readlane((int)bk, 31);
      if (__ballot(ck < kmax) == 0ull) continue;   // wave-uniform skip
      // bitonic sort ck ascending across lanes: flip = dir(lane,j) != dir(lane,k)
      CE(ck, 1, ((lane & 1) == 0) != ((lane & 2) == 0))
      CE(ck, 2, ((lane & 2) == 0) != ((lane & 4) == 0))
      CE(ck, 1, ((lane & 1) == 0) != ((lane & 4) == 0))
      CE(ck, 4, ((lane & 4) == 0) != ((lane & 8) == 0))
      CE(ck, 2, ((lane & 2) == 0) != ((lane & 8) == 0))
      CE(ck, 1, ((lane & 1) == 0) != ((lane & 8) == 0))
      CE(ck, 8, ((lane & 8) == 0) != ((lane & 16) == 0))
      CE(ck, 4, ((lane & 4) == 0) != ((lane & 16) == 0))
      CE(ck, 2, ((lane & 2) == 0) != ((lane & 16) == 0))
      CE(ck, 1, ((lane & 1) == 0) != ((lane & 16) == 0))
      CE(ck, 16, (lane & 16) != 0)   // k=32 passes: ascending everywhere
      CE(ck, 8,  (lane & 8)  != 0)
      CE(ck, 4,  (lane & 4)  != 0)
      CE(ck, 2,  (lane & 2)  != 0)
      CE(ck, 1,  (lane & 1)  != 0)
      // elementwise min with lane-reversed candidates -> bitonic 32 smallest
      { const u64 rk = SWZ64(ck, XORP(31)); if (rk < bk) bk = rk; }
      // bitonic merge back to ascending
      CE(bk, 16, (lane & 16) != 0)
      CE(bk, 8,  (lane & 8)  != 0)
      CE(bk, 4,  (lane & 4)  != 0)
      CE(bk, 2,  (lane & 2)  != 0)
      CE(bk, 1,  (lane & 1)  != 0)
    }
  }

  // ---- gather + recenter: lane == neighbor rank (ascending d2, stable ties)
  const int g  = ctile * 16 + w;
  const int bi = (int)(unsigned)bk;      // index in low 32 bits of key
  float att[A_];
#pragma unroll
  for (int c = 0; c < A_; ++c) att[c] = base[(size_t)bi * A_ + c];
#pragma unroll
  for (int c = 0; c < 3; ++c) att[c] -= sC[w][c];
  float* o = out_nei + (((size_t)(b * NG + g)) * GS + lane) * A_;
#pragma unroll
  for (int c = 0; c < A_; ++c) o[c] = att[c];
}

// ---------------------------------------------------------------------------
extern "C" void kernel_launch(void* const* d_in, const int* in_sizes, int n_in,
                              void* d_out, int out_size, void* d_ws, size_t ws_size,
                              hipStream_t stream) {
  (void)in_sizes; (void)n_in; (void)d_ws; (void)ws_size; (void)out_size;
  const float* xyz = (const float*)d_in[0];
  float* out = (float*)d_out;

  const size_t NEI  = (size_t)B_ * NG * GS * A_;  // neighborhood
  const size_t CIDX = (size_t)B_ * NG;            // center_idx (as float)
  const size_t CATT = (size_t)B_ * NG * A_;       // centroids_attrs

  float* out_nei  = out;
  float* out_idx  = out + NEI;
  float* out_attr = out + NEI + CIDX;
  float* out_coor = out + NEI + CIDX + CATT;

  fps_kernel<<<dim3(B_), dim3(1024), 0, stream>>>(xyz, out_idx, out_attr, out_coor);
  knn_group_kernel<<<dim3(NG / 16, B_), dim3(512), 0, stream>>>(xyz, out_attr, out_nei);
}